// ChebNet_65317862637698
// MI455X (gfx1250) — compile-verified
//
#include <hip/hip_runtime.h>
#include <hip/hip_bf16.h>
#include <math.h>

// ---------------------------------------------------------------------------
// ChebNet (K=5) on gfx1250: fp32 WMMA GEMMs (LDS-staged weights, 16x64 tiles
// per wave) + L2-atomic sparse propagation.
// ---------------------------------------------------------------------------

#define N_NODES 50000
#define N_EDGES 800000
#define N_GRAPHS 64
#define C_IN 128
#define C_HID 128
#define C_OUT 16

typedef __attribute__((ext_vector_type(2))) float v2f;
typedef __attribute__((ext_vector_type(8))) float v8f;

// ---------------------------------------------------------------- utilities

__global__ void zero_kernel(float* __restrict__ p, int n) {
    int i = blockIdx.x * blockDim.x + threadIdx.x;
    if (i < n) p[i] = 0.0f;
}

// dst = -src, vectorized (n must be a multiple of 4)
__global__ void negcopy4_kernel(const float* __restrict__ s, float* __restrict__ d, int n4) {
    int i = blockIdx.x * blockDim.x + threadIdx.x;
    if (i < n4) {
        float4 v = ((const float4*)s)[i];
        float4 o; o.x = -v.x; o.y = -v.y; o.z = -v.z; o.w = -v.w;
        ((float4*)d)[i] = o;
    }
}

// ---------------------------------------------------------------- edge norm

__global__ void degree_kernel(const int* __restrict__ src, float* __restrict__ deg, int n_edges) {
    int e = blockIdx.x * blockDim.x + threadIdx.x;
    if (e < n_edges) atomicAdd(&deg[src[e]], 1.0f);
}

__global__ void dinv_kernel(float* __restrict__ deg, int n_nodes) {
    int i = blockIdx.x * blockDim.x + threadIdx.x;
    if (i < n_nodes) {
        float d = deg[i];
        deg[i] = (d > 0.0f) ? rsqrtf(fmaxf(d, 1.0f)) : 0.0f;   // in-place -> dinv
    }
}

__global__ void norm_kernel(const int* __restrict__ src, const int* __restrict__ dst,
                            const float* __restrict__ dinv, float* __restrict__ norm,
                            int n_edges) {
    int e = blockIdx.x * blockDim.x + threadIdx.x;
    if (e < n_edges) norm[e] = -dinv[src[e]] * dinv[dst[e]];
}

// ------------------------------------------------------------- propagation
// P[dst[e]] += scale * norm[e] * h[src[e]]   (128-ch, 32 lanes x float4/edge)

__global__ void propagate_kernel(const float* __restrict__ h,
                                 const int* __restrict__ src,
                                 const int* __restrict__ dst,
                                 const float* __restrict__ norm,
                                 float* __restrict__ P,
                                 float scale, int n_edges) {
    int gid = blockIdx.x * blockDim.x + threadIdx.x;
    int e = gid >> 5;
    if (e >= n_edges) return;
    int c = (gid & 31) * 4;
    float s = scale * norm[e];
    const float4 hv = *(const float4*)(h + (size_t)src[e] * 128 + c);
    float* p = P + (size_t)dst[e] * 128 + c;
    atomicAdd(p + 0, s * hv.x);
    atomicAdd(p + 1, s * hv.y);
    atomicAdd(p + 2, s * hv.z);
    atomicAdd(p + 3, s * hv.w);
}

// --------------------------------------------------------------- WMMA GEMM
// OUT[n x F_OUT] (+)= A[n x 128] @ W[128 x F_OUT].
// One wave computes a 16 x (CPW*16) tile: one A fragment feeds CPW WMMAs.
// W column-group is staged in LDS (transposed, padded) once per block.
// fp32 WMMA 16x16x4 layout: A/B frag lanes 0-15 hold K={k,k+1}, lanes 16-31
// K={k+2,k+3}; C/D: VGPR v -> M = v + 8*(lane>=16), N = lane%16.

template<int F_OUT, int CPW, bool INIT, bool RELU>
__global__ void wmma_gemm_kernel(const float* __restrict__ A,
                                 const float* __restrict__ W,
                                 const float* __restrict__ bias,
                                 float* __restrict__ OUT,
                                 int n_row_tiles) {
    constexpr int CG = CPW * 16;      // columns handled per block
    constexpr int KPAD = 130;         // padded K stride (bank-conflict relief)
    __shared__ float Bl[CG * KPAD];

    const int cgBase = blockIdx.y * CG;

    // cooperative stage of W[:, cgBase : cgBase+CG] -> LDS (transposed)
    for (int idx = threadIdx.x; idx < 128 * CG; idx += blockDim.x) {
        int k = idx / CG;
        int c = idx % CG;
        Bl[c * KPAD + k] = W[(size_t)k * F_OUT + cgBase + c];
    }
    __syncthreads();

    const int rowTile = blockIdx.x * 4 + (int)(threadIdx.x >> 5);
    if (rowTile >= n_row_tiles) return;   // whole-wave uniform

    const int lane = threadIdx.x & 31;
    const int half = lane >> 4;           // selects K pair {0,1} vs {2,3}
    const int l16  = lane & 15;
    const float* arow = A + (size_t)(rowTile * 16 + l16) * 128;

    v8f acc[CPW];
    #pragma unroll
    for (int ct = 0; ct < CPW; ++ct) {
        if (INIT) {
            float bv = bias[cgBase + ct * 16 + l16];
            #pragma unroll
            for (int v = 0; v < 8; ++v) acc[ct][v] = bv;
        } else {
            #pragma unroll
            for (int v = 0; v < 8; ++v) {
                int r = rowTile * 16 + half * 8 + v;
                acc[ct][v] = OUT[(size_t)r * F_OUT + cgBase + ct * 16 + l16];
            }
        }
    }

    #pragma unroll
    for (int k = 0; k < 128; k += 4) {
        v2f a;
        a.x = arow[k + 2 * half + 0];
        a.y = arow[k + 2 * half + 1];
        #pragma unroll
        for (int ct = 0; ct < CPW; ++ct) {
            const float2 bb = *(const float2*)&Bl[(ct * 16 + l16) * KPAD + k + 2 * half];
            v2f b; b.x = bb.x; b.y = bb.y;
            acc[ct] = __builtin_amdgcn_wmma_f32_16x16x4_f32(
                          /*neg_a=*/false, a, /*neg_b=*/false, b,
                          /*c_mod=*/(short)0, acc[ct],
                          /*reuse_a=*/false, /*reuse_b=*/false);
        }
    }

    #pragma unroll
    for (int ct = 0; ct < CPW; ++ct) {
        #pragma unroll
        for (int v = 0; v < 8; ++v) {
            int r = rowTile * 16 + half * 8 + v;
            float o = acc[ct][v];
            if (RELU) o = fmaxf(o, 0.0f);
            OUT[(size_t)r * F_OUT + cgBase + ct * 16 + l16] = o;
        }
    }
}

// ------------------------------------------------------------ pooling head

__global__ void pool_kernel(const float* __restrict__ h, const int* __restrict__ batch,
                            float* __restrict__ sums, float* __restrict__ cnts, int n_nodes) {
    int i = blockIdx.x * blockDim.x + threadIdx.x;
    if (i >= n_nodes) return;
    int g = batch[i];
    const float* row = h + (size_t)i * C_OUT;
    float* srow = sums + (size_t)g * C_OUT;
    #pragma unroll
    for (int c = 0; c < C_OUT; ++c) atomicAdd(&srow[c], row[c]);
    atomicAdd(&cnts[g], 1.0f);
}

__global__ void logsoftmax_kernel(const float* __restrict__ sums,
                                  const float* __restrict__ cnts,
                                  float* __restrict__ out) {
    int g = blockIdx.x * blockDim.x + threadIdx.x;
    if (g >= N_GRAPHS) return;
    float inv = 1.0f / fmaxf(cnts[g], 1.0f);
    float v[C_OUT];
    float m = -INFINITY;
    #pragma unroll
    for (int c = 0; c < C_OUT; ++c) { v[c] = sums[g * C_OUT + c] * inv; m = fmaxf(m, v[c]); }
    float lse = 0.0f;
    #pragma unroll
    for (int c = 0; c < C_OUT; ++c) lse += __expf(v[c] - m);
    lse = __logf(lse);
    #pragma unroll
    for (int c = 0; c < C_OUT; ++c) out[g * C_OUT + c] = v[c] - m - lse;
}

// ---------------------------------------------------------------- launcher

static inline int cdiv(int a, int b) { return (a + b - 1) / b; }

extern "C" void kernel_launch(void* const* d_in, const int* in_sizes, int n_in,
                              void* d_out, int out_size, void* d_ws, size_t ws_size,
                              hipStream_t stream) {
    (void)in_sizes; (void)n_in; (void)out_size; (void)ws_size;

    const float* x     = (const float*)d_in[0];
    const int*   src   = (const int*)  d_in[1];
    const int*   dst   = (const int*)  d_in[2];
    const int*   batch = (const int*)  d_in[3];
    const float* Wt1   = (const float*)d_in[4];   // [5,128,128]
    const float* b1    = (const float*)d_in[5];
    const float* Wt2   = (const float*)d_in[6];   // [5,128,128]
    const float* b2    = (const float*)d_in[7];
    const float* Wt3   = (const float*)d_in[8];   // [5,128,16]
    const float* b3    = (const float*)d_in[9];
    float* out = (float*)d_out;

    // workspace carve-up (floats)
    const size_t NF = (size_t)N_NODES * 128;      // 6,400,000
    float* ws   = (float*)d_ws;
    float* deg  = ws;                             // 50048 (padded)
    float* nrm  = deg + 50048;                    // 800000
    float* T0   = nrm + N_EDGES;                  // big buffers x4
    float* T1   = T0 + NF;
    float* T2   = T1 + NF;
    float* T3   = T2 + NF;
    float* H3   = T3 + NF;                        // 50000*16
    float* psum = H3 + (size_t)N_NODES * C_OUT;   // 64*16
    float* pcnt = psum + N_GRAPHS * C_OUT;        // 64

    const int EB = 256;
    const int propBlocks = cdiv(N_EDGES * 32, EB);
    const int zeroBlocksNF = cdiv((int)NF, EB);
    const int neg4Blocks = cdiv((int)(NF / 4), EB);
    const int rowTiles = N_NODES / 16;            // 3125 (exact)
    const dim3 g128(cdiv(rowTiles, 4), 128 / 64); // 4 row tiles/block, 64-col groups
    const dim3 g16(cdiv(rowTiles, 4), 1);         // 16-col single group

    // ---- edge normalization ----
    zero_kernel<<<cdiv(N_NODES, EB), EB, 0, stream>>>(deg, N_NODES);
    degree_kernel<<<cdiv(N_EDGES, EB), EB, 0, stream>>>(src, deg, N_EDGES);
    dinv_kernel<<<cdiv(N_NODES, EB), EB, 0, stream>>>(deg, N_NODES);
    norm_kernel<<<cdiv(N_EDGES, EB), EB, 0, stream>>>(src, dst, deg, nrm, N_EDGES);

    // convenience macros over the fixed shapes
    #define GEMM128(A_, W_, B_, O_, INIT_, RELU_) \
        wmma_gemm_kernel<128, 4, INIT_, RELU_><<<g128, 128, 0, stream>>>(A_, W_, B_, O_, rowTiles)
    #define GEMM16(A_, W_, B_, O_, INIT_, RELU_) \
        wmma_gemm_kernel<16, 1, INIT_, RELU_><<<g16, 128, 0, stream>>>(A_, W_, B_, O_, rowTiles)
    #define PROP(H_, P_, S_) \
        propagate_kernel<<<propBlocks, EB, 0, stream>>>(H_, src, dst, nrm, P_, S_, N_EDGES)
    #define ZERO_NF(P_)  zero_kernel<<<zeroBlocksNF, EB, 0, stream>>>(P_, (int)NF)
    #define NEG_NF(S_, D_) negcopy4_kernel<<<neg4Blocks, EB, 0, stream>>>(S_, D_, (int)(NF / 4))

    const size_t WS = 128 * 128;   // stride between W[k] slices (layers 1,2)

    // ---- layer 1: in = x (const), out = T0, scratch T1,T2,T3 ----
    GEMM128(x, Wt1 + 0 * WS, b1, T0, true, false);
    ZERO_NF(T1);  PROP(x,  T1, 1.0f);
    GEMM128(T1, Wt1 + 1 * WS, b1, T0, false, false);
    NEG_NF(x,  T2);  PROP(T1, T2, 2.0f);
    GEMM128(T2, Wt1 + 2 * WS, b1, T0, false, false);
    NEG_NF(T1, T3);  PROP(T2, T3, 2.0f);
    GEMM128(T3, Wt1 + 3 * WS, b1, T0, false, false);
    NEG_NF(T2, T1);  PROP(T3, T1, 2.0f);
    GEMM128(T1, Wt1 + 4 * WS, b1, T0, false, true);          // ReLU -> h1 in T0

    // ---- layer 2: in = T0, out = T1, scratch T2,T3 (reuse T0 after k=2) ----
    GEMM128(T0, Wt2 + 0 * WS, b2, T1, true, false);
    ZERO_NF(T2);  PROP(T0, T2, 1.0f);
    GEMM128(T2, Wt2 + 1 * WS, b2, T1, false, false);
    NEG_NF(T0, T3);  PROP(T2, T3, 2.0f);
    GEMM128(T3, Wt2 + 2 * WS, b2, T1, false, false);
    NEG_NF(T2, T0);  PROP(T3, T0, 2.0f);
    GEMM128(T0, Wt2 + 3 * WS, b2, T1, false, false);
    NEG_NF(T3, T2);  PROP(T0, T2, 2.0f);
    GEMM128(T2, Wt2 + 4 * WS, b2, T1, false, true);          // ReLU -> h2 in T1

    // ---- layer 3: in = T1, out = H3 (16 ch), scratch T0,T2,T3 ----
    const size_t WS3 = 128 * 16;
    GEMM16(T1, Wt3 + 0 * WS3, b3, H3, true, false);
    ZERO_NF(T0);  PROP(T1, T0, 1.0f);
    GEMM16(T0, Wt3 + 1 * WS3, b3, H3, false, false);
    NEG_NF(T1, T2);  PROP(T0, T2, 2.0f);
    GEMM16(T2, Wt3 + 2 * WS3, b3, H3, false, false);
    NEG_NF(T0, T3);  PROP(T2, T3, 2.0f);
    GEMM16(T3, Wt3 + 3 * WS3, b3, H3, false, false);
    NEG_NF(T2, T0);  PROP(T3, T0, 2.0f);
    GEMM16(T0, Wt3 + 4 * WS3, b3, H3, false, false);

    // ---- global mean pool + log_softmax ----
    zero_kernel<<<cdiv(N_GRAPHS * C_OUT + N_GRAPHS, EB), EB, 0, stream>>>(psum, N_GRAPHS * C_OUT + N_GRAPHS);
    pool_kernel<<<cdiv(N_NODES, EB), EB, 0, stream>>>(H3, batch, psum, pcnt, N_NODES);
    logsoftmax_kernel<<<1, 64, 0, stream>>>(psum, pcnt, out);

    #undef GEMM128
    #undef GEMM16
    #undef PROP
    #undef ZERO_NF
    #undef NEG_NF
}